// BGRL_87840671138373
// MI455X (gfx1250) — compile-verified
//
#include <hip/hip_runtime.h>

// ---------------------------------------------------------------------------
// BGRL forward on MI455X (gfx1250, wave32).
//   * 8 GEMMs [100000x128]@[128x128] via V_WMMA_F32_16X16X4_F32 (fp32-exact),
//     LDS-staged A tile + transposed W (bank-conflict-free b64 fragment loads)
//   * fused 4-way SpMM edge pass, float4 gathers + hardware fp32 atomics
//   * batchnorm stats via column reduction; bn+prelu folded into A staging
// ---------------------------------------------------------------------------

constexpr int N_NODES = 100000;
constexpr int D       = 128;
constexpr int E_EDGES = 800000;
constexpr int ND      = N_NODES * D;          // 12,800,000 floats
constexpr float BN_EPS = 1e-5f;

constexpr int APAD = 132;   // A tile row pitch (floats): banks advance 4/row
constexpr int WPAD = 68;    // transposed-W row pitch (half-K=64 + 4 pad)

typedef float v2f __attribute__((ext_vector_type(2)));
typedef float v8f __attribute__((ext_vector_type(8)));

// ---------------------------------------------------------------------------
// x2 = x + perb
__global__ __launch_bounds__(256)
void add_kernel(const float* __restrict__ x, const float* __restrict__ p,
                float* __restrict__ out) {
    int i = blockIdx.x * 256 + threadIdx.x;
    out[i] = x[i] + p[i];
}

// ---------------------------------------------------------------------------
// out[M,128] = act(A[M,128]) @ W[128,128] (+ bias[n])
// Grid: 6250 blocks x 256 threads (8 waves); wave w owns tile (16 x 16) at
// (blockIdx*16, w*16). K processed in two LDS phases of 64.
// A frag (16x4 f32): lane<16 -> K={k,k+1} of row M=lane; lane>=16 -> K={k+2,k+3}.
// B frag (4x16 f32): mirrored, N=lane&15; Wlds is W transposed so both B
// k-values are one ds_load_b64.
template<bool DO_BN, bool HAS_BIAS>
__global__ __launch_bounds__(256)
void gemm128_wmma(const float* __restrict__ A, const float* __restrict__ W,
                  const float* __restrict__ bias,     // [128] (HAS_BIAS)
                  const float* __restrict__ bnscale,  // [128] (DO_BN)
                  const float* __restrict__ bnshift,  // [128] (DO_BN)
                  const float* __restrict__ prelu_a_ptr,
                  float* __restrict__ out) {
    __shared__ float Alds[16 * APAD];    //  8448 B
    __shared__ float Wlds[128 * WPAD];   // 34816 B

    const int tid  = threadIdx.x;
    const int lane = tid & 31;
    const int wave = tid >> 5;
    const int m0   = blockIdx.x * 16;
    const int half = lane >> 4;          // 0 or 1
    const int l15  = lane & 15;
    const int bn   = wave * 16 + l15;    // this lane's output column

    const float pa = DO_BN ? *prelu_a_ptr : 0.f;

    // ---- stage A tile (16 x 128), bn+prelu applied once here ----
    {
        const int i0  = tid * 8;              // 2048 floats / 256 threads
        const int row = i0 >> 7;
        const int col = i0 & 127;
        const float* src = A + (m0 + row) * D + col;
        float v[8];
        #pragma unroll
        for (int u = 0; u < 8; ++u) v[u] = src[u];
        if (DO_BN) {
            #pragma unroll
            for (int u = 0; u < 8; ++u) {
                float t = v[u] * bnscale[col + u] + bnshift[col + u];
                v[u] = t > 0.f ? t : pa * t;
            }
        }
        #pragma unroll
        for (int u = 0; u < 8; ++u) Alds[row * APAD + col + u] = v[u];
    }

    v8f acc = {};
    #pragma unroll
    for (int p = 0; p < 2; ++p) {
        __syncthreads();  // A staged (p=0) / Wlds phase-0 fully consumed (p=1)
        // ---- stage W[64p..64p+64) transposed: Wlds[n][k-64p] ----
        #pragma unroll
        for (int j = 0; j < 32; ++j) {
            const int idx = tid + 256 * j;    // 8192 floats
            const int k = idx >> 7;
            const int n = idx & 127;
            Wlds[n * WPAD + k] = W[(64 * p + k) * D + n];
        }
        __syncthreads();
        // ---- 16 WMMA steps over this K half ----
        #pragma unroll
        for (int kk = 0; kk < 64; kk += 4) {
            const int ka = kk + half * 2;
            v2f a = *(const v2f*)&Alds[l15 * APAD + 64 * p + ka];
            v2f b = *(const v2f*)&Wlds[bn * WPAD + ka];
            acc = __builtin_amdgcn_wmma_f32_16x16x4_f32(
                false, a, false, b, (short)0, acc, false, false);
        }
    }

    const float bv = HAS_BIAS ? bias[bn] : 0.f;
    // D layout: VGPR i -> row m0 + half*8 + i, column bn
    const int obase = (m0 + half * 8) * D + bn;
    #pragma unroll
    for (int i = 0; i < 8; ++i)
        out[obase + i * D] = acc[i] + bv;
}

// ---------------------------------------------------------------------------
// Fused 4-way SpMM: for each edge e, agg_j[rows[e]] += vals[e] * h_j[cols[e]]
// 32 threads (1 wave) per edge, float4 gathers, hardware fp32 global atomics.
__global__ __launch_bounds__(256)
void edge_spmm(const int* __restrict__ rows, const int* __restrict__ cols,
               const float* __restrict__ vals,
               const float* __restrict__ h0, const float* __restrict__ h1,
               const float* __restrict__ h2, const float* __restrict__ h3,
               float* __restrict__ a0, float* __restrict__ a1,
               float* __restrict__ a2, float* __restrict__ a3) {
    const int gid = blockIdx.x * 256 + threadIdx.x;
    const int e = gid >> 5;            // one wave per edge
    const int t = (gid & 31) << 2;     // 4 consecutive features per lane
    const int   r = rows[e];
    const int   c = cols[e];
    const float v = vals[e];
    const int ci = c * D + t;
    const int ri = r * D + t;
    const float4 g0 = *(const float4*)(h0 + ci);
    const float4 g1 = *(const float4*)(h1 + ci);
    const float4 g2 = *(const float4*)(h2 + ci);
    const float4 g3 = *(const float4*)(h3 + ci);
    unsafeAtomicAdd(a0 + ri + 0, v * g0.x); unsafeAtomicAdd(a0 + ri + 1, v * g0.y);
    unsafeAtomicAdd(a0 + ri + 2, v * g0.z); unsafeAtomicAdd(a0 + ri + 3, v * g0.w);
    unsafeAtomicAdd(a1 + ri + 0, v * g1.x); unsafeAtomicAdd(a1 + ri + 1, v * g1.y);
    unsafeAtomicAdd(a1 + ri + 2, v * g1.z); unsafeAtomicAdd(a1 + ri + 3, v * g1.w);
    unsafeAtomicAdd(a2 + ri + 0, v * g2.x); unsafeAtomicAdd(a2 + ri + 1, v * g2.y);
    unsafeAtomicAdd(a2 + ri + 2, v * g2.z); unsafeAtomicAdd(a2 + ri + 3, v * g2.w);
    unsafeAtomicAdd(a3 + ri + 0, v * g3.x); unsafeAtomicAdd(a3 + ri + 1, v * g3.y);
    unsafeAtomicAdd(a3 + ri + 2, v * g3.z); unsafeAtomicAdd(a3 + ri + 3, v * g3.w);
}

// ---------------------------------------------------------------------------
// c1[n] = sum_k b_on[k]*l1w[k,n] + l1b[n]   (bias folding for predictor L1)
__global__ __launch_bounds__(128)
void prep_c1(const float* __restrict__ b_on, const float* __restrict__ l1w,
             const float* __restrict__ l1b, float* __restrict__ c1) {
    int n = threadIdx.x;
    float s = l1b[n];
    for (int k = 0; k < D; ++k) s += b_on[k] * l1w[k * D + n];
    c1[n] = s;
}

// ---------------------------------------------------------------------------
// Column sums & sum-of-squares of H[100000,128] -> sums[0:128]=sum, [128:256]=sumsq
__global__ __launch_bounds__(128)
void colstats(const float* __restrict__ H, float* __restrict__ sums) {
    int col = threadIdx.x;
    float s = 0.f, q = 0.f;
    for (int r = blockIdx.x; r < N_NODES; r += gridDim.x) {
        float v = H[r * D + col];
        s += v; q += v * v;
    }
    unsafeAtomicAdd(&sums[col], s);
    unsafeAtomicAdd(&sums[128 + col], q);
}

// ---------------------------------------------------------------------------
// bn scale/shift from accumulated stats (population variance, matches jnp.var)
__global__ __launch_bounds__(128)
void bn_finalize(const float* __restrict__ sums, const float* __restrict__ bn_g,
                 const float* __restrict__ bn_b, float* __restrict__ scale,
                 float* __restrict__ shift) {
    int c = threadIdx.x;
    float mu  = sums[c] * (1.f / N_NODES);
    float var = sums[128 + c] * (1.f / N_NODES) - mu * mu;
    float sc  = bn_g[c] * rsqrtf(var + BN_EPS);
    scale[c] = sc;
    shift[c] = bn_b[c] - mu * sc;
}

// ---------------------------------------------------------------------------
// Per-node cosine loss term: 2 - 2 * <p,t>/(|p||t|), t = agg + b_tg. One block/node.
__global__ __launch_bounds__(128)
void loss_kernel(const float* __restrict__ P, const float* __restrict__ AGG,
                 const float* __restrict__ b_tg, float* __restrict__ acc) {
    __shared__ float sdp[128], spp[128], stt[128];
    const int node = blockIdx.x;
    const int t = threadIdx.x;
    float p  = P[node * D + t];
    float tv = AGG[node * D + t] + b_tg[t];
    sdp[t] = p * tv; spp[t] = p * p; stt[t] = tv * tv;
    __syncthreads();
    for (int s = 64; s > 0; s >>= 1) {
        if (t < s) { sdp[t] += sdp[t + s]; spp[t] += spp[t + s]; stt[t] += stt[t + s]; }
        __syncthreads();
    }
    if (t == 0) {
        float l = 2.f - 2.f * sdp[0] * rsqrtf(spp[0]) * rsqrtf(stt[0]);
        unsafeAtomicAdd(acc, l);
    }
}

// ---------------------------------------------------------------------------
// embed = x2 + agg_on_x2 + b_on
__global__ __launch_bounds__(256)
void embed_kernel(const float* __restrict__ x2, const float* __restrict__ agg,
                  const float* __restrict__ b_on, float* __restrict__ out) {
    int i = blockIdx.x * 256 + threadIdx.x;
    out[i] = x2[i] + agg[i] + b_on[i & (D - 1)];
}

__global__ void finalize_loss(const float* __restrict__ acc, float* __restrict__ out) {
    out[0] = acc[0] * (1.f / N_NODES);
}

// ---------------------------------------------------------------------------
extern "C" void kernel_launch(void* const* d_in, const int* in_sizes, int n_in,
                              void* d_out, int out_size, void* d_ws, size_t ws_size,
                              hipStream_t stream) {
    const float* x      = (const float*)d_in[0];
    const float* perb   = (const float*)d_in[1];
    const float* vals   = (const float*)d_in[2];
    const float* w_on   = (const float*)d_in[3];
    const float* b_on   = (const float*)d_in[4];
    const float* w_tg   = (const float*)d_in[5];
    const float* b_tg   = (const float*)d_in[6];
    const float* l1w    = (const float*)d_in[7];
    const float* l1b    = (const float*)d_in[8];
    const float* bn_g   = (const float*)d_in[9];
    const float* bn_b   = (const float*)d_in[10];
    const float* preluA = (const float*)d_in[11];
    const float* l2w    = (const float*)d_in[12];
    const float* l2b    = (const float*)d_in[13];
    const int*   rows   = (const int*)d_in[14];
    const int*   cols   = (const int*)d_in[15];
    float* out = (float*)d_out;

    float* ws = (float*)d_ws;
    float* X2        = ws + 0L * ND;
    float* H_ON_X    = ws + 1L * ND;   // reused as H1 (branch X)
    float* H_ON_X2   = ws + 2L * ND;   // reused as P  (branch X)
    float* H_TG_X    = ws + 3L * ND;   // reused as H1 (branch Y)
    float* H_TG_X2   = ws + 4L * ND;   // reused as P  (branch Y)
    float* AGG_ON_X  = ws + 5L * ND;
    float* AGG_ON_X2 = ws + 6L * ND;
    float* AGG_TG_X  = ws + 7L * ND;
    float* AGG_TG_X2 = ws + 8L * ND;
    float* STATS     = ws + 9L * ND;
    float* C1       = STATS;           // 128
    float* SUMS     = STATS + 128;     // 256 (sum | sumsq)
    float* BN_SCALE = STATS + 384;     // 128
    float* BN_SHIFT = STATS + 512;     // 128
    float* LOSS_ACC = STATS + 640;     // 1

    const int eltBlocks  = ND / 256;                 // 50000
    const int gemmBlocks = N_NODES / 16;             // 6250 (exact)
    const int edgeBlocks = (E_EDGES * 32) / 256;     // 100000

    // x2 = x + perb
    add_kernel<<<eltBlocks, 256, 0, stream>>>(x, perb, X2);

    // 4 GCN feature GEMMs (no bias / no activation)
    gemm128_wmma<false,false><<<gemmBlocks, 256, 0, stream>>>(x,  w_on, nullptr, nullptr, nullptr, preluA, H_ON_X);
    gemm128_wmma<false,false><<<gemmBlocks, 256, 0, stream>>>(X2, w_on, nullptr, nullptr, nullptr, preluA, H_ON_X2);
    gemm128_wmma<false,false><<<gemmBlocks, 256, 0, stream>>>(x,  w_tg, nullptr, nullptr, nullptr, preluA, H_TG_X);
    gemm128_wmma<false,false><<<gemmBlocks, 256, 0, stream>>>(X2, w_tg, nullptr, nullptr, nullptr, preluA, H_TG_X2);

    // zero the 4 agg buffers + stats tail (fresh every call)
    hipMemsetAsync(AGG_ON_X, 0, 4L * ND * sizeof(float), stream);
    hipMemsetAsync(STATS, 0, 1024 * sizeof(float), stream);

    prep_c1<<<1, 128, 0, stream>>>(b_on, l1w, l1b, C1);

    // fused 4-way sparse aggregation
    edge_spmm<<<edgeBlocks, 256, 0, stream>>>(rows, cols, vals,
                                              H_ON_X, H_ON_X2, H_TG_X, H_TG_X2,
                                              AGG_ON_X, AGG_ON_X2, AGG_TG_X, AGG_TG_X2);

    // embed = x2 + gcn(x2, w_on, b_on)
    embed_kernel<<<eltBlocks, 256, 0, stream>>>(X2, AGG_ON_X2, b_on, out);

    // ---- predictor branch X: px from online_x = agg_on_x + b_on ----
    gemm128_wmma<false,true><<<gemmBlocks, 256, 0, stream>>>(AGG_ON_X, l1w, C1, nullptr, nullptr, preluA, H_ON_X);
    colstats<<<512, 128, 0, stream>>>(H_ON_X, SUMS);
    bn_finalize<<<1, 128, 0, stream>>>(SUMS, bn_g, bn_b, BN_SCALE, BN_SHIFT);
    gemm128_wmma<true,true><<<gemmBlocks, 256, 0, stream>>>(H_ON_X, l2w, l2b, BN_SCALE, BN_SHIFT, preluA, H_ON_X2);
    // loss(px, target_x = gcn(x2, w_tg, b_tg))
    loss_kernel<<<N_NODES, 128, 0, stream>>>(H_ON_X2, AGG_TG_X2, b_tg, LOSS_ACC);

    // ---- predictor branch Y: py from online_y = agg_on_x2 + b_on ----
    hipMemsetAsync(SUMS, 0, 256 * sizeof(float), stream);
    gemm128_wmma<false,true><<<gemmBlocks, 256, 0, stream>>>(AGG_ON_X2, l1w, C1, nullptr, nullptr, preluA, H_TG_X);
    colstats<<<512, 128, 0, stream>>>(H_TG_X, SUMS);
    bn_finalize<<<1, 128, 0, stream>>>(SUMS, bn_g, bn_b, BN_SCALE, BN_SHIFT);
    gemm128_wmma<true,true><<<gemmBlocks, 256, 0, stream>>>(H_TG_X, l2w, l2b, BN_SCALE, BN_SHIFT, preluA, H_TG_X2);
    // loss(py, target_y = gcn(x, w_tg, b_tg))
    loss_kernel<<<N_NODES, 128, 0, stream>>>(H_TG_X2, AGG_TG_X, b_tg, LOSS_ACC);

    finalize_loss<<<1, 1, 0, stream>>>(LOSS_ACC, out + ND);
}